// BatteryCellPhy_56813827391769
// MI455X (gfx1250) — compile-verified
//
#include <hip/hip_runtime.h>
#include <cstdint>

// ---- Battery model constants (folded at compile time, double -> float) ----
#define TLEN 1024

constexpr double dR = 8.3144621, dF = 96487.0;
constexpr double dVolume = 2e-5;
constexpr double dVolS   = 0.1 * dVolume;          // 2e-6
constexpr double dVolB   = dVolume - dVolS;        // 1.8e-5
constexpr double dtD     = 7e6;
constexpr double dqmax   = 7600.0 / 0.6;
constexpr double dto = 6.08671, dtsn = 1001.38, dtsp = 46.4311;

constexpr float U0P = 4.03f, U0N = 0.01f;
constexpr float R_F    = (float)(dR / dF);
constexpr float INVF   = (float)(1.0 / dF);
constexpr float RO     = 0.117215f;
constexpr float KN     = 2120.96f;
constexpr float KP     = 248898.0f;
constexpr float INV_SN = (float)(1.0 / 0.000437545);
constexpr float INV_SP = (float)(1.0 / 0.00030962);

constexpr float VOLS = (float)dVolS, VOLB = (float)dVolB;
constexpr float INV_VOL  = (float)(1.0 / dVolume);
constexpr float INV_VOLS = (float)(1.0 / dVolS);
constexpr float INV_VOLB = (float)(1.0 / dVolB);
constexpr float INV_QSMAX = (float)(1.0 / (dqmax * dVolS / dVolume));

// delta recurrence: d' = LAMQ*d +/- (DT/VolS)*i  (DT = 1)
constexpr double dLamQ = 1.0 - (1.0 / dVolB + 1.0 / dVolS) / dtD;
constexpr float LAMQ = (float)dLamQ;
constexpr float INV_LAMQ = (float)(1.0 / dLamQ);
constexpr float CDEL = (float)(1.0 / dVolS);

// output IIRs: V' = A*V + Bc*Vnew, output uses pre-update state (exclusive scan)
constexpr double dA_O = 1.0 - 1.0 / dto, dA_SN = 1.0 - 1.0 / dtsn, dA_SP = 1.0 - 1.0 / dtsp;
constexpr float B_O  = (float)(1.0 / dto),  A_O  = (float)dA_O,  IA_O  = (float)(1.0 / dA_O);
constexpr float B_SN = (float)(1.0 / dtsn), A_SN = (float)dA_SN, IA_SN = (float)(1.0 / dA_SN);
constexpr float B_SP = (float)(1.0 / dtsp), A_SP = (float)dA_SP, IA_SP = (float)(1.0 / dA_SP);

// lam^32 via constexpr squaring chain (for the cross-wave combine)
constexpr float pow32f(double x) { double r = x; for (int i = 0; i < 5; ++i) r *= r; return (float)r; }
constexpr float LAMQ32 = pow32f(dLamQ);
constexpr float A_O32  = pow32f(dA_O);
constexpr float A_SN32 = pow32f(dA_SN);
constexpr float A_SP32 = pow32f(dA_SP);

// Intra-wave inclusive constant-coefficient affine scan (wave32 Kogge-Stone):
//   returns z_t = sum_{j<=t, in wave} lam^(t-j) * x_j
__device__ __forceinline__ float wave_iir_scan(float x, float lam, int lane) {
    float v = x, lp = lam;
    #pragma unroll
    for (int off = 1; off < 32; off <<= 1) {
        float up = __shfl_up(v, off, 32);
        if (lane >= off) v = fmaf(lp, up, v);
        lp = lp * lp;
    }
    return v;
}

// asinh for strictly positive O(1) arguments (same formula the reference expands to)
__device__ __forceinline__ float asinh_pos(float u) {
    return __logf(u + __fsqrt_rn(fmaf(u, u, 1.0f)));
}

// Redlich-Kister: (b*P(b) - 2x(1-x)*P'(b)) / F, P(b)=sum A_k b^k, dual Horner.
__device__ __forceinline__ float vint13(float x, const float a[13]) {
    float b = fmaf(2.0f, x, -1.0f);
    float P = a[12], dP = 0.0f;
    #pragma unroll
    for (int k = 11; k >= 0; --k) {
        dP = fmaf(dP, b, P);
        P  = fmaf(P, b, a[k]);
    }
    float r = (b * P - 2.0f * x * (1.0f - x) * dP) * INVF;
    return (x != 0.5f) ? r : 0.0f;       // reference's x==0.5 guard
}

__global__ __launch_bounds__(TLEN, 1)
void battery_scan_kernel(const float* __restrict__ i_in,
                         const float* __restrict__ x0,
                         const float* __restrict__ Aps,
                         const float* __restrict__ Ans,
                         float* __restrict__ out) {
    const int t = threadIdx.x;            // time step handled by this thread
    const int b = blockIdx.x;             // cell handled by this workgroup
    const int lane = t & 31, wave = t >> 5;

    __shared__ float lds_i[TLEN];
    __shared__ float partA[2][32];        // input-scan partials
    __shared__ float partC[3][32];        // output-scan partials (distinct: no WAR hazard)

    // --- CDNA5 async data mover: stage this cell's current trace into LDS ---
    // (ASYNCcnt path). Issue first; overlap the DMA with the scalar loads below.
    const float* gsrc = i_in + (size_t)b * (size_t)TLEN;   // uniform base -> SGPR pair
    {
        unsigned gOff = (unsigned)t * 4u;                   // per-lane byte offset
        unsigned ldsAddr = (unsigned)(uintptr_t)(&lds_i[t]); // low 32 bits = LDS offset
        asm volatile("global_load_async_to_lds_b32 %0, %1, %2"
                     :: "v"(ldsAddr), "v"(gOff), "s"(gsrc) : "memory");
    }

    // per-cell initial state + RK coefficients (uniform -> scalar loads, overlap DMA)
    const float tb   = x0[b * 8 + 0];
    const float Vo0  = x0[b * 8 + 1];
    const float Vsn0 = x0[b * 8 + 2];
    const float Vsp0 = x0[b * 8 + 3];
    const float qnB0 = x0[b * 8 + 4], qnS0 = x0[b * 8 + 5];
    const float qpB0 = x0[b * 8 + 6], qpS0 = x0[b * 8 + 7];
    float a[13];
    #pragma unroll
    for (int k = 0; k < 13; ++k) a[k] = Aps[k];
    const float an0 = Ans[0];

    asm volatile("s_wait_asynccnt 0" ::: "memory");
    __syncthreads();
    const float iv = lds_i[t];

    // --- Phase 1: two input scans (exclusive: state at step t uses j < t) ---
    float zS = wave_iir_scan(iv, 1.0f, lane);   // prefix sum of i
    float zW = wave_iir_scan(iv, LAMQ, lane);   // IIR of i (lambda = LAMQ)
    if (lane == 31) { partA[0][wave] = zS; partA[1][wave] = zW; }
    __syncthreads();
    if (wave == 0) {                            // batched cross-wave combine
        float p0 = partA[0][lane], p1 = partA[1][lane];
        float m1 = LAMQ32;
        #pragma unroll
        for (int off = 1; off < 32; off <<= 1) {
            float u0 = __shfl_up(p0, off, 32);
            float u1 = __shfl_up(p1, off, 32);
            if (lane >= off) { p0 = p0 + u0; p1 = fmaf(m1, u1, p1); }
            m1 = m1 * m1;
        }
        partA[0][lane] = p0; partA[1][lane] = p1;
    }
    __syncthreads();
    {
        const float c0 = (wave == 0) ? 0.0f : partA[0][wave - 1];
        const float c1 = (wave == 0) ? 0.0f : partA[1][wave - 1];
        zS = zS + c0;
        zW = fmaf(__powf(LAMQ, (float)(lane + 1)), c1, zW);
    }
    const float S = zS - iv;                    // exclusive prefix sum
    const float W = (zW - iv) * INV_LAMQ;       // exclusive IIR

    // --- reconstruct q-state at step t (sigma/delta coordinates, exact linear map) ---
    const float sn0 = qnB0 + qnS0, sp0 = qpB0 + qpS0;
    const float dn0 = qnB0 * INV_VOLB - qnS0 * INV_VOLS;   // 0 for reference x0
    const float dp0 = qpB0 * INV_VOLB - qpS0 * INV_VOLS;

    const float lamT = __powf(LAMQ, (float)t);
    const float dn = fmaf( CDEL, W, lamT * dn0);
    const float dp = fmaf(-CDEL, W, lamT * dp0);
    const float sn = sn0 - S;                    // DT = 1
    const float sp = sp0 + S;

    const float qnS = VOLS * (sn - VOLB * dn) * INV_VOL;
    const float qpS = VOLS * (sp - VOLB * dp) * INV_VOL;
    const float xnS = qnS * INV_QSMAX;
    const float xpS = qpS * INV_QSMAX;

    // --- Phase 2: pointwise nonlinear electrochemistry (fully parallel) ---
    const float RTF = R_F * tb;
    const float Jn  = iv * INV_SN;
    const float Jp  = iv * INV_SP;
    const float Jn0 = KN * __fsqrt_rn(xnS * (1.0f - xnS));   // alpha = 0.5
    const float Jp0 = KP * __fsqrt_rn(xpS * (1.0f - xpS));
    const float VsnN = 2.0f * RTF * asinh_pos(Jn / (2.0f * Jn0));
    const float VspN = 2.0f * RTF * asinh_pos(Jp / (2.0f * Jp0));
    const float bn = fmaf(2.0f, xnS, -1.0f);
    const float vintN = (xnS != 0.5f) ? bn * an0 * INVF : 0.0f;  // 1-term RK
    const float Ven = U0N + RTF * __logf((1.0f - xnS) / xnS) + vintN;
    const float Vep = U0P + RTF * __logf((1.0f - xpS) / xpS) + vint13(xpS, a);
    const float VoN = iv * RO;

    // --- Phase 3: three output IIRs (exclusive: v_t uses pre-update V) ---
    const float uo = B_O  * VoN;
    const float un = B_SN * VsnN;
    const float up = B_SP * VspN;
    float zo = wave_iir_scan(uo, A_O,  lane);
    float zn = wave_iir_scan(un, A_SN, lane);
    float zp = wave_iir_scan(up, A_SP, lane);
    if (lane == 31) { partC[0][wave] = zo; partC[1][wave] = zn; partC[2][wave] = zp; }
    __syncthreads();
    if (wave == 0) {                            // batched cross-wave combine (3 scans)
        float p0 = partC[0][lane], p1 = partC[1][lane], p2 = partC[2][lane];
        float m0 = A_O32, m1 = A_SN32, m2 = A_SP32;
        #pragma unroll
        for (int off = 1; off < 32; off <<= 1) {
            float u0 = __shfl_up(p0, off, 32);
            float u1 = __shfl_up(p1, off, 32);
            float u2 = __shfl_up(p2, off, 32);
            if (lane >= off) {
                p0 = fmaf(m0, u0, p0);
                p1 = fmaf(m1, u1, p1);
                p2 = fmaf(m2, u2, p2);
            }
            m0 = m0 * m0; m1 = m1 * m1; m2 = m2 * m2;
        }
        partC[0][lane] = p0; partC[1][lane] = p1; partC[2][lane] = p2;
    }
    __syncthreads();
    {
        const float c0 = (wave == 0) ? 0.0f : partC[0][wave - 1];
        const float c1 = (wave == 0) ? 0.0f : partC[1][wave - 1];
        const float c2 = (wave == 0) ? 0.0f : partC[2][wave - 1];
        const float fl = (float)(lane + 1);
        zo = fmaf(__powf(A_O,  fl), c0, zo);
        zn = fmaf(__powf(A_SN, fl), c1, zn);
        zp = fmaf(__powf(A_SP, fl), c2, zp);
    }
    const float tf  = (float)t;
    const float Vo  = fmaf(__powf(A_O,  tf), Vo0,  (zo - uo) * IA_O);
    const float Vsn = fmaf(__powf(A_SN, tf), Vsn0, (zn - un) * IA_SN);
    const float Vsp = fmaf(__powf(A_SP, tf), Vsp0, (zp - up) * IA_SP);

    out[(size_t)b * TLEN + t] = Vep - Ven - Vo - Vsn - Vsp;
}

extern "C" void kernel_launch(void* const* d_in, const int* in_sizes, int n_in,
                              void* d_out, int out_size, void* d_ws, size_t ws_size,
                              hipStream_t stream) {
    const float* i_in = (const float*)d_in[0];   // [B, T] fp32
    const float* x0   = (const float*)d_in[1];   // [B, 8] fp32
    const float* Aps  = (const float*)d_in[2];   // [13]  fp32
    const float* Ans  = (const float*)d_in[3];   // [1]   fp32
    float* out = (float*)d_out;                  // [B, T] fp32

    const int B = in_sizes[0] / TLEN;            // 2048 cells
    battery_scan_kernel<<<dim3(B), dim3(TLEN), 0, stream>>>(i_in, x0, Aps, Ans, out);
}